// ActiveShiftLayer_57363583205651
// MI455X (gfx1250) — compile-verified
//
#include <hip/hip_runtime.h>

#define N_  32
#define C_  256
#define H_  56
#define W_  56
#define PLANE   (H_ * W_)          /* 3136 floats */
#define CHUNKS  (PLANE / 4)        /* 784 x 16-byte chunks (rows never straddle: 56%4==0) */
#define ROWV    (W_ / 4)           /* 14 vec4 per row */
#define BLOCK   256

typedef int   v4i __attribute__((vector_size(16)));
typedef float v4f __attribute__((ext_vector_type(4)));

__global__ __launch_bounds__(BLOCK)
void ActiveShiftLayer_57363583205651_kernel(const float* __restrict__ x,
                                            const float* __restrict__ shift,
                                            float* __restrict__ out) {
    __shared__ float sm[PLANE];

    const int p   = blockIdx.x;            // plane index = n*C + c
    const int c   = p & (C_ - 1);          // C = 256 (power of two)
    const int tid = threadIdx.x;

    const float* plane_in  = x   + (size_t)p * PLANE;
    float*       plane_out = out + (size_t)p * PLANE;

    // ---- Stage the whole 56x56 plane into LDS (gfx1250 async copy path) ----
#if defined(__HIP_DEVICE_COMPILE__) && __has_builtin(__builtin_amdgcn_global_load_async_to_lds_b128)
    for (int chunk = tid; chunk < CHUNKS; chunk += BLOCK) {
        v4i* g = (v4i*)(plane_in + chunk * 4);
        v4i* l = (v4i*)(sm + chunk * 4);
        __builtin_amdgcn_global_load_async_to_lds_b128(g, l, 0, 0);
    }
#if __has_builtin(__builtin_amdgcn_s_wait_asynccnt)
    __builtin_amdgcn_s_wait_asynccnt(0);
#else
    asm volatile("s_wait_asynccnt 0" ::: "memory");
#endif
#else
    // Fallback: vector copy through registers (still single HBM read per elem)
    for (int chunk = tid; chunk < CHUNKS; chunk += BLOCK) {
        const v4f v = *reinterpret_cast<const v4f*>(plane_in + chunk * 4);
        *reinterpret_cast<v4f*>(sm + chunk * 4) = v;
    }
#endif
    __syncthreads();

    // ---- Per-channel shift parameters (uniform across the block) ----
    const float alpha = shift[2 * c + 0];
    const float beta  = shift[2 * c + 1];
    const float fa = floorf(alpha);
    const float fb = floorf(beta);
    const float ta = alpha - fa;
    const float tb = beta  - fb;
    const int   ia = (int)fa;
    const int   ib = (int)fb;

    const float w00 = (1.f - ta) * (1.f - tb);
    const float w01 = (1.f - ta) * tb;
    const float w10 = ta * (1.f - tb);
    const float w11 = ta * tb;

    // ---- Bilinear blend, 4 outputs per iteration; 10 LDS reads per chunk ----
    for (int chunk = tid; chunk < CHUNKS; chunk += BLOCK) {
        const int h  = chunk / ROWV;               // 0..55
        const int wb = (chunk - h * ROWV) * 4;     // 0,4,...,52

        const int h0 = h + ia, h1 = h0 + 1;
        const bool h0v = (unsigned)h0 < (unsigned)H_;
        const bool h1v = (unsigned)h1 < (unsigned)H_;
        const int h0o = min(max(h0, 0), H_ - 1) * W_;
        const int h1o = min(max(h1, 0), H_ - 1) * W_;

        float r0[5], r1[5];
#pragma unroll
        for (int j = 0; j < 5; ++j) {
            const int  wj = wb + j + ib;
            const bool wv = (unsigned)wj < (unsigned)W_;
            const int  wc = min(max(wj, 0), W_ - 1);
            r0[j] = (h0v && wv) ? sm[h0o + wc] : 0.f;
            r1[j] = (h1v && wv) ? sm[h1o + wc] : 0.f;
        }

        v4f o;
        o.x = w00 * r0[0] + w01 * r0[1] + w10 * r1[0] + w11 * r1[1];
        o.y = w00 * r0[1] + w01 * r0[2] + w10 * r1[1] + w11 * r1[2];
        o.z = w00 * r0[2] + w01 * r0[3] + w10 * r1[2] + w11 * r1[3];
        o.w = w00 * r0[3] + w01 * r0[4] + w10 * r1[3] + w11 * r1[4];

        __builtin_nontemporal_store(o, reinterpret_cast<v4f*>(plane_out + chunk * 4));
    }
}

extern "C" void kernel_launch(void* const* d_in, const int* in_sizes, int n_in,
                              void* d_out, int out_size, void* d_ws, size_t ws_size,
                              hipStream_t stream) {
    const float* x     = (const float*)d_in[0];   // (32, 256, 56, 56) fp32
    const float* shift = (const float*)d_in[1];   // (256, 2) fp32
    float*       out   = (float*)d_out;           // same shape as x

    const int planes = N_ * C_;                   // 8192 workgroups
    ActiveShiftLayer_57363583205651_kernel<<<dim3(planes), dim3(BLOCK), 0, stream>>>(
        x, shift, out);
}